// GAT_layer_66700842107292
// MI455X (gfx1250) — compile-verified
//
#include <hip/hip_runtime.h>

// ---------------------------------------------------------------------------
// GAT layer for MI455X (gfx1250).
//   N_NODES=100000, N_EDGES=1600000, F_IN=128, H=4, F_OUT=32 (H*F_OUT=128)
// Pipeline: zero -> WMMA fp32 GEMM (+bias) -> node scores -> edge exp/denom
//           (global_atomic_add_f32) -> weighted scatter-add into d_out.
// ---------------------------------------------------------------------------

#define N_NODES 100000
#define N_EDGES 1600000
#define FDIM    128          // F_IN == N_HEADS*F_OUT == 128
#define NHEADS  4

typedef float v2f __attribute__((ext_vector_type(2)));
typedef float v8f __attribute__((ext_vector_type(8)));

// Hardware f32 atomic add at device scope (output region is L2-resident).
__device__ __forceinline__ void atomic_add_f32_dev(float* addr, float val) {
    asm volatile("global_atomic_add_f32 %0, %1, off scope:SCOPE_DEV"
                 :: "v"(addr), "v"(val) : "memory");
}

// ---------------------------------------------------------------------------
// Kernel 0: zero a float buffer (d_out accumulators + denom) every launch.
// ---------------------------------------------------------------------------
__global__ void gat_zero(float* __restrict__ p, int n) {
    int i = blockIdx.x * blockDim.x + threadIdx.x;
    if (i < n) p[i] = 0.0f;
}

// ---------------------------------------------------------------------------
// Kernel 1: h_proj = h_in @ W + b via V_WMMA_F32_16X16X4_F32.
// Block = 256 threads = 8 waves. Block computes 16 rows x 128 cols.
// Wave w owns output cols [16w, 16w+16). K = 128 -> 32 chained WMMAs.
// A tile (16x128) staged in LDS, padded stride 132 (bank-conflict-free).
// B fragments (128x16 slice of W) preloaded into VGPRs (64 regs/lane).
//
// f32 16x16x4 WMMA lane layout (wave32):
//   A (16x4): lane l holds M=l%16, K = 2*(l/16)+{0,1} in 2 VGPRs
//   B (4x16): lane l holds N=l%16, K = 2*(l/16)+{0,1} in 2 VGPRs
//   C/D (16x16): VGPR r, lane l -> M = 8*(l/16)+r, N = l%16
// ---------------------------------------------------------------------------
__global__ void __launch_bounds__(256)
gat_gemm_wmma(const float* __restrict__ h_in, const float* __restrict__ W,
              const float* __restrict__ bias, float* __restrict__ h_proj) {
    __shared__ __align__(16) float As[16 * 132];

    const int t    = threadIdx.x;
    const int wave = t >> 5;
    const int lane = t & 31;
    const int row0 = blockIdx.x * 16;          // 6250 * 16 == 100000 exactly

    // Cooperative A-tile load: 2048 floats, 8 per thread as two float4s.
    {
        const int r = t >> 4;                  // 0..15
        const int c = (t & 15) * 8;            // 0..120 step 8
        const float4* g =
            reinterpret_cast<const float4*>(h_in + (size_t)(row0 + r) * FDIM + c);
        float4 x0 = g[0];
        float4 x1 = g[1];
        *reinterpret_cast<float4*>(&As[r * 132 + c])     = x0;
        *reinterpret_cast<float4*>(&As[r * 132 + c + 4]) = x1;
    }
    __syncthreads();

    const int col  = wave * 16 + (lane & 15);  // global output column 0..127
    const int ksel = (lane >> 4) * 2;          // 0 or 2

    // Preload B fragments: W[:, col] pairs for every K-step (W is L2-hot).
    v2f bfrag[32];
#pragma unroll
    for (int kk = 0; kk < 32; ++kk) {
        const int k = kk * 4 + ksel;
        bfrag[kk].x = W[(size_t)k * FDIM + col];
        bfrag[kk].y = W[(size_t)(k + 1) * FDIM + col];
    }

    v8f acc = {};
#pragma unroll
    for (int kk = 0; kk < 32; ++kk) {
        const int k = kk * 4 + ksel;
        v2f a = *reinterpret_cast<const v2f*>(&As[(lane & 15) * 132 + k]);
        acc = __builtin_amdgcn_wmma_f32_16x16x4_f32(
            /*neg_a=*/false, a, /*neg_b=*/false, bfrag[kk],
            /*c_mod=*/(short)0, acc, /*reuse_a=*/false, /*reuse_b=*/false);
    }

    const float bv = bias[col];                // same col for all 8 C VGPRs
    const int mbase = (lane >> 4) * 8;
#pragma unroll
    for (int r = 0; r < 8; ++r) {
        h_proj[(size_t)(row0 + mbase + r) * FDIM + col] = acc[r] + bv;
    }
}

// ---------------------------------------------------------------------------
// Kernel 2: per-(node,head) score contributions.
//   score_src[n,h] = dot(h_proj[n, h*32 : h*32+32], a_src[h])
//   score_tgt[n,h] = dot(h_proj[n, h*32 : h*32+32], a_tgt[h])
// ---------------------------------------------------------------------------
__global__ void gat_scores(const float* __restrict__ hp,
                           const float* __restrict__ a_src,
                           const float* __restrict__ a_tgt,
                           float* __restrict__ ss, float* __restrict__ st,
                           int total) {
    int t = blockIdx.x * blockDim.x + threadIdx.x;
    if (t >= total) return;
    const int n = t >> 2;
    const int h = t & 3;
    const float4* row = reinterpret_cast<const float4*>(hp + (size_t)n * FDIM + h * 32);
    const float4* as  = reinterpret_cast<const float4*>(a_src + h * 32);
    const float4* at  = reinterpret_cast<const float4*>(a_tgt + h * 32);
    float s0 = 0.0f, s1 = 0.0f;
#pragma unroll
    for (int j = 0; j < 8; ++j) {
        float4 v = row[j], x = as[j], y = at[j];
        s0 += v.x * x.x + v.y * x.y + v.z * x.z + v.w * x.w;
        s1 += v.x * y.x + v.y * y.y + v.z * y.z + v.w * y.w;
    }
    ss[t] = s0;
    st[t] = s1;
}

// ---------------------------------------------------------------------------
// Kernel 3: per edge: e = LeakyReLU(score_tgt[tgt]+score_src[src], 0.2),
//           exp_e = exp(e) (stored), denom[tgt] += exp_e (atomic).
// ---------------------------------------------------------------------------
__global__ void gat_edge_exp(const int* __restrict__ ei,
                             const float* __restrict__ ss,
                             const float* __restrict__ st,
                             float* __restrict__ exp_e,
                             float* __restrict__ denom) {
    int e = blockIdx.x * blockDim.x + threadIdx.x;   // 6250*256 == N_EDGES
    const int src = ei[e];
    const int tgt = ei[N_EDGES + e];
#pragma unroll
    for (int h = 0; h < NHEADS; ++h) {
        float v = st[tgt * NHEADS + h] + ss[src * NHEADS + h];
        v = (v > 0.0f) ? v : 0.2f * v;               // LeakyReLU(0.2)
        float ex = __expf(v);                        // raw exp, as reference
        exp_e[e * NHEADS + h] = ex;
        atomic_add_f32_dev(&denom[tgt * NHEADS + h], ex);
    }
}

// ---------------------------------------------------------------------------
// Kernel 4: one wave per edge; lane l scatters features 4l..4l+3.
//   col = h*32 + (l%8)*4 == 4*l, so the source-row read is one coalesced
//   float4 per lane (512B/wave), followed by 4 L2-resident f32 atomics.
// ---------------------------------------------------------------------------
__global__ void __launch_bounds__(256)
gat_aggregate(const int* __restrict__ ei, const float* __restrict__ hp,
              const float* __restrict__ exp_e, const float* __restrict__ denom,
              float* __restrict__ out) {
    const int gid  = blockIdx.x * blockDim.x + threadIdx.x;
    const int e    = gid >> 5;                       // 200000*8 == N_EDGES
    const int lane = gid & 31;
    const int src  = ei[e];
    const int tgt  = ei[N_EDGES + e];
    const int h    = lane >> 3;                      // head for this lane

    const float alpha = exp_e[e * NHEADS + h] /
                        (denom[tgt * NHEADS + h] + 1e-16f);

    const float4 hv =
        *reinterpret_cast<const float4*>(hp + (size_t)src * FDIM + lane * 4);
    float* o = out + (size_t)tgt * FDIM + lane * 4;
    atomic_add_f32_dev(o + 0, alpha * hv.x);
    atomic_add_f32_dev(o + 1, alpha * hv.y);
    atomic_add_f32_dev(o + 2, alpha * hv.z);
    atomic_add_f32_dev(o + 3, alpha * hv.w);
}

// ---------------------------------------------------------------------------
// Launch wiring. Workspace layout (floats):
//   h_proj   : [0,            12,800,000)
//   scoreSrc : [12,800,000,   13,200,000)
//   scoreTgt : [13,200,000,   13,600,000)
//   denom    : [13,600,000,   14,000,000)
//   exp_e    : [14,000,000,   20,400,000)   (~81.6 MB total)
// ---------------------------------------------------------------------------
extern "C" void kernel_launch(void* const* d_in, const int* in_sizes, int n_in,
                              void* d_out, int out_size, void* d_ws, size_t ws_size,
                              hipStream_t stream) {
    const float* h_in  = (const float*)d_in[0];
    const int*   ei    = (const int*)d_in[1];    // edge_index, harness int mapping
    const float* W     = (const float*)d_in[2];
    const float* bias  = (const float*)d_in[3];
    const float* a_src = (const float*)d_in[4];
    const float* a_tgt = (const float*)d_in[5];
    float* out = (float*)d_out;

    float* ws      = (float*)d_ws;
    float* h_proj  = ws;
    float* ss      = ws + (size_t)12800000;
    float* st      = ss + (size_t)400000;
    float* denom   = st + (size_t)400000;
    float* exp_e   = denom + (size_t)400000;

    const int nOut    = N_NODES * FDIM;          // 12,800,000
    const int nScores = N_NODES * NHEADS;        // 400,000

    // 0) zero the accumulators (every call: deterministic).
    gat_zero<<<(nOut + 255) / 256, 256, 0, stream>>>(out, nOut);
    gat_zero<<<(nScores + 255) / 256, 256, 0, stream>>>(denom, nScores);

    // 1) projection GEMM + bias (WMMA fp32): 100000/16 = 6250 blocks.
    gat_gemm_wmma<<<N_NODES / 16, 256, 0, stream>>>(h_in, W, bias, h_proj);

    // 2) per-node attention scores.
    gat_scores<<<(nScores + 255) / 256, 256, 0, stream>>>(h_proj, a_src, a_tgt,
                                                          ss, st, nScores);

    // 3) edge exp + denominator accumulation.
    gat_edge_exp<<<N_EDGES / 256, 256, 0, stream>>>(ei, ss, st, exp_e, denom);

    // 4) weighted message scatter-add: 1 wave/edge -> E*32 threads.
    gat_aggregate<<<(N_EDGES * 32) / 256, 256, 0, stream>>>(ei, h_proj, exp_e,
                                                            denom, out);
}